// CenterLossLayer_42099269436115
// MI455X (gfx1250) — compile-verified
//
#include <hip/hip_runtime.h>
#include <hip/hip_bf16.h>
#include <stdint.h>

#define NUM_CLASSES 7000
#define FEAT_DIM    128
#define ALPHA_F     0.5f

typedef __attribute__((ext_vector_type(2))) float v2f;
typedef __attribute__((ext_vector_type(8))) float v8f;
typedef int v4i_gcc __attribute__((__vector_size__(4 * sizeof(int))));

#define AS_GLOBAL __attribute__((address_space(1)))
#define AS_LDS    __attribute__((address_space(3)))

#if defined(__gfx1250__) && __has_builtin(__builtin_amdgcn_global_load_async_to_lds_b128)
#define HAVE_ASYNC_LDS 1
#else
#define HAVE_ASYNC_LDS 0
#endif

// ---------------------------------------------------------------------------
// Wave-level (32-lane) sum reduction using V_WMMA_F32_16X16X4_F32.
// A (16x4 f32, 2 VGPRs/lane): VGPR0 = partial (K=0 lanes 0-15 / K=2 lanes 16-31),
// VGPR1 = 0. B = all ones (layout-independent). D[m][n] = p_m + p_{m+16}.
// Each lane sums its 8 accumulator VGPRs, then one xor-16 shuffle finishes.
// ---------------------------------------------------------------------------
__device__ __forceinline__ float wave_sum_wmma(float partial) {
  v2f a; a[0] = partial; a[1] = 0.0f;
  v2f b; b[0] = 1.0f;    b[1] = 1.0f;
  v8f c = {};
  c = __builtin_amdgcn_wmma_f32_16x16x4_f32(
      /*neg_a=*/false, a, /*neg_b=*/false, b,
      /*c_mod=*/(short)0, c, /*reuse_a=*/false, /*reuse_b=*/false);
  float t = c[0] + c[1] + c[2] + c[3] + c[4] + c[5] + c[6] + c[7];
  t += __shfl_xor(t, 16, 32);  // lanes 0-15 hold rows 0-7, lanes 16-31 rows 8-15
  return t;
}

#if HAVE_ASYNC_LDS
__device__ __forceinline__ void async_16B_to_lds(const float* gsrc, float* ldst) {
  __builtin_amdgcn_global_load_async_to_lds_b128(
      (AS_GLOBAL v4i_gcc*)(uintptr_t)gsrc, (AS_LDS v4i_gcc*)(uintptr_t)ldst,
      /*offset=*/0, /*cpol=*/0);
}
__device__ __forceinline__ void wait_async0() { __builtin_amdgcn_s_wait_asynccnt(0); }
__device__ __forceinline__ void wait_async1() { __builtin_amdgcn_s_wait_asynccnt(1); }
#endif

// ---------------------------------------------------------------------------
// Kernel 0: zero counts, zero loss slot, copy centers into the output buffer.
// ---------------------------------------------------------------------------
__global__ void cl_init_kernel(const float* __restrict__ centers,
                               float* __restrict__ out, int* __restrict__ counts,
                               int total_out) {
  const int tid    = blockIdx.x * blockDim.x + threadIdx.x;
  const int stride = gridDim.x * blockDim.x;
  if (tid < NUM_CLASSES) counts[tid] = 0;
  for (int i = tid; i < total_out; i += stride)
    out[i] = (i == 0) ? 0.0f : centers[i - 1];
}

// ---------------------------------------------------------------------------
// Kernel 1: per-class label histogram.
// ---------------------------------------------------------------------------
__global__ void cl_hist_kernel(const int* __restrict__ labels,
                               int* __restrict__ counts, int n) {
  const int i = blockIdx.x * blockDim.x + threadIdx.x;
  if (i < n) atomicAdd(&counts[labels[i]], 1);
}

// ---------------------------------------------------------------------------
// Kernel 2: one wave per row. Lane l covers floats [4l, 4l+4) of the 128-wide
// row. Features staged through LDS via async loads (double buffered); centers
// gathered from L2; delta^2 accumulated per lane; scatter-add via f32 atomics.
// ---------------------------------------------------------------------------
__global__ void __launch_bounds__(256)
cl_main_kernel(const float* __restrict__ features,
               const int* __restrict__ labels,
               const float* __restrict__ centers,
               const int* __restrict__ counts,
               float* __restrict__ loss_out,
               float* __restrict__ centers_out,
               int n, int total_waves) {
  __shared__ __align__(16) float lds_buf[8 * 2 * FEAT_DIM];  // 8 waves x 2 bufs x 512B
  const int lane      = threadIdx.x & 31;
  const int waveInBlk = threadIdx.x >> 5;
  const int waveId    = blockIdx.x * 8 + waveInBlk;
  float*    myLds     = &lds_buf[waveInBlk * 2 * FEAT_DIM];

  float partial = 0.0f;

#if HAVE_ASYNC_LDS
  int buf = 0;
  if (waveId < n)
    async_16B_to_lds(features + (size_t)waveId * FEAT_DIM + lane * 4,
                     myLds + lane * 4);
  for (int row = waveId; row < n; row += total_waves) {
    const int  nrow      = row + total_waves;
    const bool have_next = nrow < n;
    if (have_next)
      async_16B_to_lds(features + (size_t)nrow * FEAT_DIM + lane * 4,
                       myLds + (buf ^ 1) * FEAT_DIM + lane * 4);
    if (have_next) wait_async1(); else wait_async0();
    const float4 fv = *(const float4*)&myLds[buf * FEAT_DIM + lane * 4];
#else
  for (int row = waveId; row < n; row += total_waves) {
    const float4 fv = ((const float4*)(features + (size_t)row * FEAT_DIM))[lane];
#endif
    const int lbl = __builtin_amdgcn_readfirstlane(labels[row]);
    const float4 cv = ((const float4*)(centers + (size_t)lbl * FEAT_DIM))[lane];

    const float dx = fv.x - cv.x, dy = fv.y - cv.y;
    const float dz = fv.z - cv.z, dw = fv.w - cv.w;
    partial += dx * dx + dy * dy + dz * dz + dw * dw;

    const float cnt   = (float)__builtin_amdgcn_readfirstlane(counts[lbl]);
    const float scale = ALPHA_F / (1.0f + cnt);
    float* orow = centers_out + (size_t)lbl * FEAT_DIM + lane * 4;
    atomicAdd(orow + 0, dx * scale);
    atomicAdd(orow + 1, dy * scale);
    atomicAdd(orow + 2, dz * scale);
    atomicAdd(orow + 3, dw * scale);
#if HAVE_ASYNC_LDS
    buf ^= 1;
#endif
  }

  // Uniform control flow here: EXEC is all-1s for the WMMA reduction.
  const float s = wave_sum_wmma(partial);
  if (lane == 0) atomicAdd(loss_out, 0.5f * s);
}

// ---------------------------------------------------------------------------
extern "C" void kernel_launch(void* const* d_in, const int* in_sizes, int n_in,
                              void* d_out, int out_size, void* d_ws, size_t ws_size,
                              hipStream_t stream) {
  const float* features = (const float*)d_in[0];
  const int*   labels   = (const int*)d_in[1];
  const float* centers  = (const float*)d_in[2];
  float*       out      = (float*)d_out;
  int*         counts   = (int*)d_ws;  // NUM_CLASSES ints of scratch
  const int    n        = in_sizes[1]; // N = number of rows / labels

  cl_init_kernel<<<2048, 256, 0, stream>>>(centers, out, counts, out_size);
  cl_hist_kernel<<<(n + 255) / 256, 256, 0, stream>>>(labels, counts, n);

  const int BLOCKS = 2048;                 // 8 waves/block -> 16384 waves
  cl_main_kernel<<<BLOCKS, 256, 0, stream>>>(features, labels, centers, counts,
                                             out, out + 1, n, BLOCKS * 8);
}